// HybonetConv_56788057588100
// MI455X (gfx1250) — compile-verified
//
#include <hip/hip_runtime.h>
#include <math.h>

typedef __attribute__((ext_vector_type(2))) float v2f;
typedef __attribute__((ext_vector_type(4))) float v4f;
typedef __attribute__((ext_vector_type(8))) float v8f;

#define NROWS 8192
#define D     128
#define BM    64
#define BN    64
#define LDP   132   // padded LDS stride for 128-wide tiles (16B aligned rows, no bank conflict)
#define LDA   68    // padded LDS stride for 64-wide att tile

// D = A(16x4) * B(4x16) + C, fp32 WMMA (wave32)
__device__ __forceinline__ v8f wmma4(v2f a, v2f b, v8f c) {
  return __builtin_amdgcn_wmma_f32_16x16x4_f32(
      /*neg_a=*/false, a, /*neg_b=*/false, b,
      /*c_mod=*/(short)0, c, /*reuse_a=*/false, /*reuse_b=*/false);
}

__device__ __forceinline__ float sigmoidf_(float x) {
  return 1.0f / (1.0f + expf(-x));
}

// Load R rows x 128 cols (row-major, stride 128) from global into LDS with stride LDP.
template<int R>
__device__ __forceinline__ void load_tile(const float* __restrict__ g, float* s, int tid) {
  #pragma unroll
  for (int i = tid; i < R * 32; i += 128) {
    const int r = i >> 5;
    const int c = (i & 31) << 2;
    v4f v = *(const v4f*)(g + (size_t)r * D + c);
    *(v4f*)(s + r * LDP + c) = v;
  }
}

// ---------------------------------------------------------------------------
// y = X @ W^T + b, then Lorentz projection:
//   time = sigmoid(y0)*exp(log_scale)+1.1
//   out  = [time, y_space * sqrt((time^2-1)/clip(|y_space|^2,1e-8))]
// One block: 64 rows, 4 waves, each wave owns 16 output rows x 128 cols.
// ---------------------------------------------------------------------------
__global__ __launch_bounds__(128) void lorentz_linear_kernel(
    const float* __restrict__ X, const float* __restrict__ W,
    const float* __restrict__ bias, const float* __restrict__ log_scale,
    float* __restrict__ Y)
{
  __shared__ float Xs[BM * LDP];   // X tile (reused for y afterwards)
  __shared__ float Ws[D * LDP];    // W[o][k]
  const int tid  = threadIdx.x;
  const int lane = tid & 31;
  const int wave = tid >> 5;
  const int hi   = lane >> 4;      // 0/1 half of the wave
  const int ln   = lane & 15;
  const int i0   = blockIdx.x * BM;
  const int m0   = wave * 16;

  load_tile<BM>(X + (size_t)i0 * D, Xs, tid);
  load_tile<D >(W, Ws, tid);
  __syncthreads();

  v8f acc[8] = {};
  #pragma unroll 4
  for (int kk = 0; kk < D; kk += 4) {
    const int kb = kk + (hi << 1);
    v2f a = *(const v2f*)(Xs + (m0 + ln) * LDP + kb);       // A[m][k]
    #pragma unroll
    for (int s = 0; s < 8; ++s) {
      v2f b = *(const v2f*)(Ws + (ln + s * 16) * LDP + kb); // B[k][n] = W[n][k]
      acc[s] = wmma4(a, b, acc[s]);
    }
  }
  __syncthreads();  // done reading Xs; reuse it for y

  #pragma unroll
  for (int s = 0; s < 8; ++s) {
    #pragma unroll
    for (int v = 0; v < 8; ++v) {
      const int m = m0 + v + hi * 8;
      const int n = ln + s * 16;
      Xs[m * LDP + n] = acc[s][v] + bias[n];
    }
  }
  __syncthreads();

  if (tid < BM) {
    const float es = expf(log_scale[0]);
    const float* yr = Xs + tid * LDP;
    const float t = sigmoidf_(yr[0]) * es + 1.1f;
    float sq = 0.0f;
    #pragma unroll 4
    for (int d = 1; d < D; ++d) sq = fmaf(yr[d], yr[d], sq);
    sq = fmaxf(sq, 1e-8f);
    const float sc = sqrtf((t * t - 1.0f) / sq);
    float* o = Y + (size_t)(i0 + tid) * D;
    o[0] = t;
    #pragma unroll 4
    for (int d = 1; d < D; ++d) o[d] = yr[d] * sc;
  }
}

// ---------------------------------------------------------------------------
// Fused attention:
//   cin  = dot128(q_i, k_j) - 2*q_i[0]*k_j[0]
//   att  = adj * sigmoid((2 + 2*cin)/att_scale + att_bias)
//   support_i = sum_j att[i,j] * h[j,:]          (accumulated in VGPRs)
//   out_i = support_i / sqrt(clip(|sum s_d^2 - s_0^2|, 1e-6))
// Never materializes the 256MB att matrix; adj is streamed exactly once (NT).
// ---------------------------------------------------------------------------
__global__ __launch_bounds__(128) void lorentz_att_kernel(
    const float* __restrict__ Q, const float* __restrict__ K,
    const float* __restrict__ H, const float* __restrict__ adj,
    const float* __restrict__ att_bias_p, const float* __restrict__ att_scale_p,
    float* __restrict__ out)
{
  __shared__ float Qs[BM * LDP];
  __shared__ float Ks[BN * LDP];
  __shared__ float Hs[BN * LDP];
  __shared__ float As[BM * LDA];

  const int tid  = threadIdx.x;
  const int lane = tid & 31;
  const int wave = tid >> 5;
  const int hi   = lane >> 4;
  const int ln   = lane & 15;
  const int i0   = blockIdx.x * BM;
  const int m0   = wave * 16;

  const float bias      = att_bias_p[0];
  const float inv_scale = 1.0f / att_scale_p[0];

  load_tile<BM>(Q + (size_t)i0 * D, Qs, tid);
  v8f sacc[8] = {};

  for (int j0 = 0; j0 < NROWS; j0 += BN) {
    __syncthreads();  // previous iteration done reading Ks/Hs/As (and first-iter Qs fence)
    load_tile<BN>(K + (size_t)j0 * D, Ks, tid);
    load_tile<BN>(H + (size_t)j0 * D, Hs, tid);

    // Early, non-temporal adj loads for this tile (independent of LDS); the
    // HBM latency of the 256MB adj stream overlaps GEMM1's 128 WMMAs.
    float adjv[4][8];
    #pragma unroll
    for (int s = 0; s < 4; ++s) {
      #pragma unroll
      for (int v = 0; v < 8; ++v) {
        const int m = m0 + v + hi * 8;
        const int n = ln + s * 16;
        adjv[s][v] = __builtin_nontemporal_load(
            adj + (size_t)(i0 + m) * NROWS + (j0 + n));
      }
    }

    // Prefetch the next contiguous 32KB K/H tiles toward the WGP while this
    // tile computes (global_prefetch_b8).
    if (j0 + BN < NROWS) {
      const char* nk = (const char*)(K + (size_t)(j0 + BN) * D);
      const char* nh = (const char*)(H + (size_t)(j0 + BN) * D);
      const int off = tid * 256;
      __builtin_prefetch(nk + off, 0, 3);
      __builtin_prefetch(nk + off + 128, 0, 3);
      __builtin_prefetch(nh + off, 0, 3);
      __builtin_prefetch(nh + off + 128, 0, 3);
    }
    __syncthreads();

    // ---- GEMM1: cin tile rows [m0,m0+16) x cols [0,64), K=128 ----
    v8f cacc[4] = {};
    #pragma unroll 4
    for (int kk = 0; kk < D; kk += 4) {
      const int kb = kk + (hi << 1);
      v2f a = *(const v2f*)(Qs + (m0 + ln) * LDP + kb);       // q rows
      #pragma unroll
      for (int s = 0; s < 4; ++s) {
        v2f b = *(const v2f*)(Ks + (ln + s * 16) * LDP + kb); // B[k][n] = K[n][k]
        cacc[s] = wmma4(a, b, cacc[s]);
      }
    }

    // ---- epilogue: time fixup + sigmoid + adj mask -> As ----
    #pragma unroll
    for (int s = 0; s < 4; ++s) {
      #pragma unroll
      for (int v = 0; v < 8; ++v) {
        const int m = m0 + v + hi * 8;        // local row in [0,64)
        const int n = ln + s * 16;            // local col in [0,64)
        const float cin = cacc[s][v] - 2.0f * Qs[m * LDP] * Ks[n * LDP];
        const float z   = fmaf(2.0f + 2.0f * cin, inv_scale, bias);
        As[m * LDA + n] = adjv[s][v] * sigmoidf_(z);
      }
    }
    __syncthreads();

    // ---- GEMM2: support(16x128) += att(16x64) @ h(64x128) ----
    #pragma unroll 4
    for (int kk = 0; kk < BN; kk += 4) {
      const int kb = kk + (hi << 1);
      v2f a = *(const v2f*)(As + (m0 + ln) * LDA + kb);       // att rows (A layout)
      #pragma unroll
      for (int s = 0; s < 8; ++s) {
        v2f b;
        b.x = Hs[(kb + 0) * LDP + ln + s * 16];               // B[k][n] = h[k][n]
        b.y = Hs[(kb + 1) * LDP + ln + s * 16];
        sacc[s] = wmma4(a, b, sacc[s]);
      }
    }
  }

  __syncthreads();  // everyone finished GEMM2 reads; reuse Qs for support
  #pragma unroll
  for (int s = 0; s < 8; ++s) {
    #pragma unroll
    for (int v = 0; v < 8; ++v) {
      const int m = m0 + v + hi * 8;
      const int n = ln + s * 16;
      Qs[m * LDP + n] = sacc[s][v];
    }
  }
  __syncthreads();

  if (tid < BM) {
    const float* sr = Qs + tid * LDP;
    const float s0 = sr[0];
    float ss = 0.0f;
    #pragma unroll 4
    for (int d = 1; d < D; ++d) ss = fmaf(sr[d], sr[d], ss);
    const float li    = ss - s0 * s0;                    // l_inner
    const float denom = sqrtf(fmaxf(fabsf(li), 1e-6f)); // C == 1.0
    const float inv   = 1.0f / denom;
    float* o = out + (size_t)(i0 + tid) * D;
    #pragma unroll 4
    for (int d = 0; d < D; ++d)
      __builtin_nontemporal_store(sr[d] * inv, o + d);   // write-once output
  }
}

// ---------------------------------------------------------------------------
extern "C" void kernel_launch(void* const* d_in, const int* in_sizes, int n_in,
                              void* d_out, int out_size, void* d_ws, size_t ws_size,
                              hipStream_t stream) {
  const float* x         = (const float*)d_in[0];
  const float* adj       = (const float*)d_in[1];
  const float* W_lin     = (const float*)d_in[2];
  const float* b_lin     = (const float*)d_in[3];
  const float* scale_lin = (const float*)d_in[4];
  const float* W_q       = (const float*)d_in[5];
  const float* b_q       = (const float*)d_in[6];
  const float* scale_q   = (const float*)d_in[7];
  const float* W_k       = (const float*)d_in[8];
  const float* b_k       = (const float*)d_in[9];
  const float* scale_k   = (const float*)d_in[10];
  const float* att_bias  = (const float*)d_in[11];
  const float* att_scale = (const float*)d_in[12];
  float* out = (float*)d_out;

  float* h = (float*)d_ws;                 // 8192*128 f32 = 4 MB
  float* q = h + (size_t)NROWS * D;        // 4 MB
  float* k = q + (size_t)NROWS * D;        // 4 MB

  dim3 grid(NROWS / BM);
  dim3 block(128);
  lorentz_linear_kernel<<<grid, block, 0, stream>>>(x, W_lin, b_lin, scale_lin, h);
  lorentz_linear_kernel<<<grid, block, 0, stream>>>(h, W_q, b_q, scale_q, q);
  lorentz_linear_kernel<<<grid, block, 0, stream>>>(h, W_k, b_k, scale_k, k);
  lorentz_att_kernel  <<<grid, block, 0, stream>>>(q, k, h, adj, att_bias, att_scale, out);
}